// GCN_43112881717445
// MI455X (gfx1250) — compile-verified
//
#include <hip/hip_runtime.h>

typedef __attribute__((ext_vector_type(2))) float v2f;
typedef __attribute__((ext_vector_type(8))) float v8f;

// ---------------------------------------------------------------------------
// 1. init: deg = 1.0 (self loop), zero agg buffer and final output buffer
// ---------------------------------------------------------------------------
__global__ __launch_bounds__(256) void gcn_init_kernel(float* __restrict__ agg,
                                                       float* __restrict__ out,
                                                       float* __restrict__ deg,
                                                       int n_nodes, long long total) {
    long long i = (long long)blockIdx.x * blockDim.x + threadIdx.x;
    if (i < total) {
        agg[i] = 0.0f;
        out[i] = 0.0f;
    }
    if (i < n_nodes) deg[i] = 1.0f;  // self-loop contributes 1 to every degree
}

// ---------------------------------------------------------------------------
// 2. weight transpose: Wt[n*K + k] = W[k*N + n]   (N = 64 fixed)
//    Makes per-lane B-operand pairs contiguous -> b64 loads in the GEMM.
// ---------------------------------------------------------------------------
__global__ __launch_bounds__(256) void gcn_transpose_w_kernel(const float* __restrict__ W,
                                                              float* __restrict__ Wt,
                                                              int K) {
    int i = blockIdx.x * blockDim.x + threadIdx.x;  // over K*64 elements
    if (i < K * 64) {
        int k = i >> 6;
        int n = i & 63;
        Wt[n * K + k] = W[k * 64 + n];
    }
}

// ---------------------------------------------------------------------------
// 3. degree: deg[col[e]] += 1  (hardware f32 atomic)
// ---------------------------------------------------------------------------
__global__ __launch_bounds__(256) void gcn_degree_kernel(const long long* __restrict__ colIdx,
                                                         float* __restrict__ deg,
                                                         long long E) {
    long long e = (long long)blockIdx.x * blockDim.x + threadIdx.x;
    if (e < E) {
        unsafeAtomicAdd(&deg[(int)colIdx[e]], 1.0f);
    }
}

// ---------------------------------------------------------------------------
// 4. dinv = rsqrt(deg)   (deg >= 1 always, thanks to self loops)
// ---------------------------------------------------------------------------
__global__ __launch_bounds__(256) void gcn_rsqrt_kernel(float* __restrict__ deg, int n) {
    int i = blockIdx.x * blockDim.x + threadIdx.x;
    if (i < n) deg[i] = rsqrtf(deg[i]);
}

// ---------------------------------------------------------------------------
// fp32 WMMA GEMM:  D[nrows x 64] = A[nrows x K] @ Wt^T  (Wt is 64 x K)
//   block = 128 threads (4 waves). Block covers 16 rows; wave w covers
//   cols [16w, 16w+16). K templated -> fully unrolled chain of
//   V_WMMA_F32_16X16X4_F32 with clustered b64 loads.
//
//   A operand layout (ISA 7.12.2, 32-bit A 16x4): lane L -> M = L%16,
//   VGPR0 = K0+2*(L/16), VGPR1 = K0+2*(L/16)+1.  B mirrored on N.
//   C/D layout: VGPR r -> M = r + 8*(L/16), N = L%16.
//
//   Out-of-range rows are CLAMPED (not predicated): garbage flows only into
//   D rows that the store guard drops, so loads stay unconditional and the
//   compiler emits no exec-mask juggling in the hot loop.
// ---------------------------------------------------------------------------
template <int K>
__global__ __launch_bounds__(128) void gcn_gemm_wmma_kernel(const float* __restrict__ A,
                                                            const float* __restrict__ Wt,
                                                            float* __restrict__ D,
                                                            int nrows) {
    const int lane  = threadIdx.x & 31;
    const int wave  = threadIdx.x >> 5;   // 0..3 -> N tile
    const int laneM = lane & 15;
    const int grp   = lane >> 4;          // 0 or 1 (K split within VGPR pair)
    const int m0    = blockIdx.x * 16;

    int arow = m0 + laneM;
    arow = (arow < nrows) ? arow : (nrows - 1);   // clamp, never predicate
    const int ncol = wave * 16 + laneM;

    const v2f* ap = (const v2f*)(A  + (long long)arow * K + 2 * grp);  // stride 2 v2f / 4 floats
    const v2f* bp = (const v2f*)(Wt + (long long)ncol * K + 2 * grp);

    v8f acc = {};
#pragma unroll
    for (int k0 = 0; k0 < K; k0 += 4) {
        v2f a = ap[k0 >> 1];   // A[arow][k0 + 2*grp .. +1]
        v2f b = bp[k0 >> 1];   // Wt[ncol][k0 + 2*grp .. +1] == B[k][ncol]
        // D = A*B + C  (8 args: neg_a, A, neg_b, B, c_mod, C, reuse_a, reuse_b)
        acc = __builtin_amdgcn_wmma_f32_16x16x4_f32(
            false, a, false, b, (short)0, acc, false, false);
    }

    const int mbase = m0 + grp * 8;
#pragma unroll
    for (int r = 0; r < 8; ++r) {
        int m = mbase + r;
        if (m < nrows) D[(long long)m * 64 + ncol] = acc[r];
    }
}

// ---------------------------------------------------------------------------
// 5/8. edge scatter: one wave32 per edge; lane handles features 2L, 2L+1.
//      agg[col] += h[row] * dinv[row] * dinv[col]
//      Index loads are wave-uniform; h rows are L2-resident (25.6 MB << 192 MB).
// ---------------------------------------------------------------------------
__global__ __launch_bounds__(256) void gcn_scatter_kernel(const float* __restrict__ h,
                                                          const long long* __restrict__ rowIdx,
                                                          const long long* __restrict__ colIdx,
                                                          const float* __restrict__ dinv,
                                                          float* __restrict__ agg,
                                                          long long E) {
    long long wid = ((long long)blockIdx.x * blockDim.x + threadIdx.x) >> 5;
    int lane = threadIdx.x & 31;
    if (wid >= E) return;

    int r = (int)rowIdx[wid];
    int c = (int)colIdx[wid];
    float nrm = dinv[r] * dinv[c];

    const float2* hp = (const float2*)(h + (long long)r * 64);
    float2 v = hp[lane];

    float* dst = agg + (long long)c * 64 + lane * 2;
    unsafeAtomicAdd(dst,     v.x * nrm);
    unsafeAtomicAdd(dst + 1, v.y * nrm);
}

// ---------------------------------------------------------------------------
// 6. layer-1 combine: h = relu(agg + h * dinv^2 + b1)   (self-loop folded in)
// ---------------------------------------------------------------------------
__global__ __launch_bounds__(256) void gcn_combine_relu_kernel(const float* __restrict__ agg,
                                                               float* __restrict__ h,
                                                               const float* __restrict__ dinv,
                                                               const float* __restrict__ bias,
                                                               long long total) {
    long long i = (long long)blockIdx.x * blockDim.x + threadIdx.x;
    if (i >= total) return;
    int node = (int)(i >> 6);
    int f    = (int)(i & 63);
    float s  = dinv[node];
    float t  = agg[i] + h[i] * s * s + bias[f];
    h[i] = fmaxf(t, 0.0f);
}

// ---------------------------------------------------------------------------
// 9. layer-2 combine: out = out + h2 * dinv^2 + b2   (no relu)
// ---------------------------------------------------------------------------
__global__ __launch_bounds__(256) void gcn_combine_out_kernel(const float* __restrict__ h2,
                                                              float* __restrict__ out,
                                                              const float* __restrict__ dinv,
                                                              const float* __restrict__ bias,
                                                              long long total) {
    long long i = (long long)blockIdx.x * blockDim.x + threadIdx.x;
    if (i >= total) return;
    int node = (int)(i >> 6);
    int f    = (int)(i & 63);
    float s  = dinv[node];
    out[i] = out[i] + h2[i] * s * s + bias[f];
}

// ---------------------------------------------------------------------------
extern "C" void kernel_launch(void* const* d_in, const int* in_sizes, int n_in,
                              void* d_out, int out_size, void* d_ws, size_t ws_size,
                              hipStream_t stream) {
    const float*     x  = (const float*)d_in[0];
    const long long* ei = (const long long*)d_in[1];   // int64 [2, E]
    const float*     W1 = (const float*)d_in[2];       // [128, 64]
    const float*     b1 = (const float*)d_in[3];       // [64]
    const float*     W2 = (const float*)d_in[4];       // [64, 64]
    const float*     b2 = (const float*)d_in[5];       // [64]
    float* out = (float*)d_out;

    const int       in_ch   = 128;
    const int       F       = 64;
    const int       n_nodes = in_sizes[0] / in_ch;                 // 100000
    const long long E       = (long long)in_sizes[1] / 2;          // 1,600,000
    const long long total   = (long long)n_nodes * F;              // 6.4M

    // workspace layout: dinv | Wt1 (64x128) | Wt2 (64x64) | hbuf | aggbuf
    float* dinv = (float*)d_ws;
    size_t off  = (((size_t)n_nodes) + 255) & ~(size_t)255;
    float* Wt1  = dinv + off;           off += (size_t)F * in_ch;  // 8192
    float* Wt2  = dinv + off;           off += (size_t)F * F;      // 4096
    float* hbuf   = dinv + off;         off += (size_t)total;
    float* aggbuf = dinv + off;

    const long long* rowIdx = ei;       // edge_index[0] (source)
    const long long* colIdx = ei + E;   // edge_index[1] (target)

    const int B = 256;
    int gTotal = (int)((total + B - 1) / B);
    int gEdgeT = (int)((E + B - 1) / B);
    int gNodes = (n_nodes + B - 1) / B;
    int gGemm  = (n_nodes + 15) / 16;
    int gScat  = (int)((E + 7) / 8);    // 8 waves (edges) per block

    // 1. zero agg + out, deg = 1
    gcn_init_kernel<<<gTotal, B, 0, stream>>>(aggbuf, out, dinv, n_nodes, total);
    // 2. transpose weights (tiny)
    gcn_transpose_w_kernel<<<(in_ch * 64 + B - 1) / B, B, 0, stream>>>(W1, Wt1, in_ch);
    gcn_transpose_w_kernel<<<(F * 64 + B - 1) / B, B, 0, stream>>>(W2, Wt2, F);
    // 3. degree accumulation
    gcn_degree_kernel<<<gEdgeT, B, 0, stream>>>(colIdx, dinv, E);
    // 4. dinv = rsqrt(deg)
    gcn_rsqrt_kernel<<<gNodes, B, 0, stream>>>(dinv, n_nodes);
    // 5. h = x @ W1 (fp32 WMMA, fully unrolled K=128)
    gcn_gemm_wmma_kernel<128><<<gGemm, 128, 0, stream>>>(x, Wt1, hbuf, n_nodes);
    // 6. agg = scatter(h * norm)
    gcn_scatter_kernel<<<gScat, B, 0, stream>>>(hbuf, rowIdx, colIdx, dinv, aggbuf, E);
    // 7. h = relu(agg + h*dinv^2 + b1)   (in place)
    gcn_combine_relu_kernel<<<gTotal, B, 0, stream>>>(aggbuf, hbuf, dinv, b1, total);
    // 8. h2 = h @ W2 (fp32 WMMA, fully unrolled K=64) -> reuse aggbuf
    gcn_gemm_wmma_kernel<64><<<gGemm, 128, 0, stream>>>(hbuf, Wt2, aggbuf, n_nodes);
    // 9. out = scatter(h2 * norm)   (out zeroed in step 1)
    gcn_scatter_kernel<<<gScat, B, 0, stream>>>(aggbuf, rowIdx, colIdx, dinv, out, E);
    // 10. out += h2*dinv^2 + b2
    gcn_combine_out_kernel<<<gTotal, B, 0, stream>>>(aggbuf, out, dinv, b2, total);
}